// QSVDDModel_44684839748154
// MI455X (gfx1250) — compile-verified
//
#include <hip/hip_runtime.h>
#include <math.h>

// ---------------------------------------------------------------------------
// Types
// ---------------------------------------------------------------------------
typedef __attribute__((ext_vector_type(2))) float v2f;
typedef __attribute__((ext_vector_type(8))) float v8f;

struct cpx { float x, y; };

__device__ __forceinline__ cpx cmul(cpx a, cpx b) {
    return { a.x * b.x - a.y * b.y, a.x * b.y + a.y * b.x };
}
__device__ __forceinline__ cpx cadd(cpx a, cpx b) { return { a.x + b.x, a.y + b.y }; }

// ---------------------------------------------------------------------------
// Kernel A: build the 5 distinct 4x4 SU(4) gate matrices from weights.
// G = K2 * C01 * (RY(p8) x I) * C10 * (RY(p6) x RZ(p7)) * C01 * K1
// (basis index = 2*b_w0 + b_w1 ; C01 = CNOT ctrl MSB, C10 = CNOT ctrl LSB)
// ---------------------------------------------------------------------------
__device__ void u3m(float t, float p, float l, cpx U[2][2]) {
    float c = cosf(0.5f * t), s = sinf(0.5f * t);
    float sp, cp, sl, cl;
    sincosf(p, &sp, &cp);
    sincosf(l, &sl, &cl);
    float cpl = cp * cl - sp * sl;
    float spl = sp * cl + cp * sl;
    U[0][0] = { c, 0.0f };
    U[0][1] = { -cl * s, -sl * s };     // -e^{il} s
    U[1][0] = { cp * s, sp * s };       //  e^{ip} s
    U[1][1] = { cpl * c, spl * c };     //  e^{i(p+l)} c
}

__device__ void kron22(const cpx A[2][2], const cpx B[2][2], cpx O[4][4]) {
    for (int a0 = 0; a0 < 2; ++a0)
        for (int b0 = 0; b0 < 2; ++b0)
            for (int a1 = 0; a1 < 2; ++a1)
                for (int b1 = 0; b1 < 2; ++b1)
                    O[2 * a0 + b0][2 * a1 + b1] = cmul(A[a0][a1], B[b0][b1]);
}

__device__ void mul44(const cpx A[4][4], const cpx B[4][4], cpx O[4][4]) {
    for (int r = 0; r < 4; ++r)
        for (int c = 0; c < 4; ++c) {
            cpx s = { 0.0f, 0.0f };
            for (int k = 0; k < 4; ++k) s = cadd(s, cmul(A[r][k], B[k][c]));
            O[r][c] = s;
        }
}

__device__ void swaprows(cpx M[4][4], int r0, int r1) {
    for (int c = 0; c < 4; ++c) { cpx t = M[r0][c]; M[r0][c] = M[r1][c]; M[r1][c] = t; }
}

__global__ void build_gates(const float* __restrict__ w, cpx* __restrict__ G) {
    int i = threadIdx.x;
    if (i >= 5) return;
    const float* p = w + i * 15;

    cpx A[2][2], B[2][2], M[4][4], T[4][4], K[4][4];

    // K1 = U3(p0,p1,p2) (w0) ⊗ U3(p3,p4,p5) (w1)
    u3m(p[0], p[1], p[2], A);
    u3m(p[3], p[4], p[5], B);
    kron22(A, B, M);
    // CNOT(ctrl=w0): swap rows 2,3
    swaprows(M, 2, 3);
    // RY(p6) on w0 ⊗ RZ(p7) on w1
    {
        float c = cosf(0.5f * p[6]), s = sinf(0.5f * p[6]);
        A[0][0] = { c, 0 }; A[0][1] = { -s, 0 }; A[1][0] = { s, 0 }; A[1][1] = { c, 0 };
        float cz = cosf(0.5f * p[7]), sz = sinf(0.5f * p[7]);
        B[0][0] = { cz, -sz }; B[0][1] = { 0, 0 }; B[1][0] = { 0, 0 }; B[1][1] = { cz, sz };
        kron22(A, B, T);
        mul44(T, M, K);
        for (int r = 0; r < 4; ++r) for (int c2 = 0; c2 < 4; ++c2) M[r][c2] = K[r][c2];
    }
    // CNOT(ctrl=w1): swap rows 1,3
    swaprows(M, 1, 3);
    // RY(p8) on w0 ⊗ I
    {
        float c = cosf(0.5f * p[8]), s = sinf(0.5f * p[8]);
        A[0][0] = { c, 0 }; A[0][1] = { -s, 0 }; A[1][0] = { s, 0 }; A[1][1] = { c, 0 };
        B[0][0] = { 1, 0 }; B[0][1] = { 0, 0 }; B[1][0] = { 0, 0 }; B[1][1] = { 1, 0 };
        kron22(A, B, T);
        mul44(T, M, K);
        for (int r = 0; r < 4; ++r) for (int c2 = 0; c2 < 4; ++c2) M[r][c2] = K[r][c2];
    }
    // CNOT(ctrl=w0): swap rows 2,3
    swaprows(M, 2, 3);
    // K2 = U3(p9,p10,p11) ⊗ U3(p12,p13,p14)
    u3m(p[9], p[10], p[11], A);
    u3m(p[12], p[13], p[14], B);
    kron22(A, B, T);
    mul44(T, M, K);

    for (int r = 0; r < 4; ++r)
        for (int c2 = 0; c2 < 4; ++c2)
            G[i * 16 + r * 4 + c2] = K[r][c2];
}

// ---------------------------------------------------------------------------
// Kernel B: build U (256x256, row-major re/im planes) column by column.
// Block j evolves basis state e_j through the 25 two-qubit gates in LDS.
// Qubit q maps to bit (7-q) of the flat index (qubit 0 = MSB).
// ---------------------------------------------------------------------------
__global__ __launch_bounds__(64) void build_unitary(const cpx* __restrict__ G,
                                                    float* __restrict__ Ure,
                                                    float* __restrict__ Uim) {
    __shared__ cpx st[256];
    __shared__ cpx Gs[5][4][4];
    const int t = threadIdx.x;      // 64 threads
    const int col = blockIdx.x;     // column of U

    for (int i = t; i < 80; i += 64) ((cpx*)Gs)[i] = G[i];
    for (int i = t; i < 256; i += 64) st[i] = { (i == col) ? 1.0f : 0.0f, 0.0f };
    __syncthreads();

    const int gl[25]  = {0,0,0,0,0,0,0,0, 1,1,1,1,1,1,1,1, 2,2,2,2, 3,3,3,3, 4};
    const int gw0[25] = {0,2,4,6,1,3,5,7, 0,2,4,6,1,3,5,7, 2,6,0,4, 2,6,0,4, 2};
    const int gw1[25] = {1,3,5,7,2,4,6,0, 1,3,5,7,2,4,6,0, 4,0,2,6, 4,0,2,6, 6};

    for (int g = 0; g < 25; ++g) {
        const int p0 = 7 - gw0[g], p1 = 7 - gw1[g];
        const int m0 = 1 << p0, m1 = 1 << p1;
        // scatter the 6 free bits of t into positions != p0,p1
        int base = 0, gg = t;
        for (int pos = 0; pos < 8; ++pos) {
            if (pos == p0 || pos == p1) continue;
            base |= (gg & 1) << pos;
            gg >>= 1;
        }
        cpx v0 = st[base], v1 = st[base | m1], v2 = st[base | m0], v3 = st[base | m0 | m1];
        const cpx(*Gm)[4] = Gs[gl[g]];
        cpx r[4];
        for (int k = 0; k < 4; ++k)
            r[k] = cadd(cadd(cmul(Gm[k][0], v0), cmul(Gm[k][1], v1)),
                        cadd(cmul(Gm[k][2], v2), cmul(Gm[k][3], v3)));
        st[base] = r[0]; st[base | m1] = r[1]; st[base | m0] = r[2]; st[base | m0 | m1] = r[3];
        __syncthreads();
    }

    for (int i = t; i < 256; i += 64) {
        Ure[i * 256 + col] = st[i].x;
        Uim[i * 256 + col] = st[i].y;
    }
}

// ---------------------------------------------------------------------------
// Kernel C: Phi = X * U^T via V_WMMA_F32_16X16X4_F32, fused Pauli measurement.
// 512 blocks x 256 threads (8 wave32). Each block: 16 batch rows, all 256 cols.
// ---------------------------------------------------------------------------
#define LDSW 258   // padded row length (floats) to break bank conflicts

__global__ __launch_bounds__(256) void gemm_measure(const float* __restrict__ x,
                                                    const float* __restrict__ Ure,
                                                    const float* __restrict__ Uim,
                                                    float* __restrict__ out) {
    __shared__ float As[16][LDSW];
    __shared__ float Pre[16][LDSW];
    __shared__ float Pim[16][LDSW];
    __shared__ float ssq[16];

    const int t = threadIdx.x;
    const int b0 = blockIdx.x * 16;

    // Stage 16x256 X tile (coalesced)
    for (int i = t; i < 4096; i += 256)
        As[i >> 8][i & 255] = x[(b0 + (i >> 8)) * 256 + (i & 255)];
    __syncthreads();

    // Per-row |x|^2 (normalization folded into the final expectations)
    if (t < 16) {
        float s = 0.0f;
        for (int c = 0; c < 256; ++c) { float v = As[t][c]; s += v * v; }
        ssq[t] = s;
    }

    const int wave = t >> 5, lane = t & 31;
    const int m = lane & 15, half = lane >> 4;

    // Each wave: two 16-column tiles of Phi (re & im accumulators)
    for (int ct = 0; ct < 2; ++ct) {
        const int c0 = wave * 32 + ct * 16;
        const int brow = c0 + m;                 // B fragment N index -> U row
        v8f accRe = {0, 0, 0, 0, 0, 0, 0, 0};
        v8f accIm = {0, 0, 0, 0, 0, 0, 0, 0};

        for (int k0 = 0; k0 < 256; k0 += 4) {
            const int kk = k0 + 2 * half;
            // A: 16x4 f32, lane -> M, half selects K pair (ISA A-layout)
            v2f a = *(const v2f*)&As[m][kk];
            // B: 4x16 f32, lane -> N, VGPR+half -> K (ISA B-layout); L2-resident U
            v2f br = *(const v2f*)(Ure + brow * 256 + kk);
            v2f bi = *(const v2f*)(Uim + brow * 256 + kk);
            accRe = __builtin_amdgcn_wmma_f32_16x16x4_f32(false, a, false, br,
                                                          (short)0, accRe, false, false);
            accIm = __builtin_amdgcn_wmma_f32_16x16x4_f32(false, a, false, bi,
                                                          (short)0, accIm, false, false);
        }
        // D layout: VGPR v -> M = v + 8*half, lane -> N
        for (int v = 0; v < 8; ++v) {
            const int mm = v + 8 * half;
            Pre[mm][c0 + m] = accRe[v];
            Pim[mm][c0 + m] = accIm[v];
        }
    }
    __syncthreads();

    // Measurement: qubit 2 -> bit 5 (mask 32), qubit 6 -> bit 1 (mask 2).
    // Order: X2, Y2, Z2, X6, Y6, Z6, X2X6, Y2Y6, Z2Z6
    for (int si = 0; si < 2; ++si) {
        const int s = wave * 2 + si;
        float e[9];
        for (int k = 0; k < 9; ++k) e[k] = 0.0f;

        for (int i = lane; i < 256; i += 32) {
            const float ar = Pre[s][i], ai = Pim[s][i];
            const int j2 = i ^ 32, j6 = i ^ 2, j26 = i ^ 34;
            const float br2 = Pre[s][j2],  bi2 = Pim[s][j2];
            const float br6 = Pre[s][j6],  bi6 = Pim[s][j6];
            const float brb = Pre[s][j26], bib = Pim[s][j26];
            const float s2 = ((i >> 5) & 1) ? -1.0f : 1.0f;
            const float s6 = ((i >> 1) & 1) ? -1.0f : 1.0f;
            const float n2 = ar * ar + ai * ai;

            e[0] += ar * br2 + ai * bi2;              // X2: Re(conj(a)*b)
            e[1] += -s2 * (ai * br2 - ar * bi2);      // Y2: Re(conj(a)*i*(2b2-1)*b)
            e[2] += s2 * n2;                          // Z2
            e[3] += ar * br6 + ai * bi6;              // X6
            e[4] += -s6 * (ai * br6 - ar * bi6);      // Y6
            e[5] += s6 * n2;                          // Z6
            e[6] += ar * brb + ai * bib;              // X2X6
            e[7] += -s2 * s6 * (ar * brb + ai * bib); // Y2Y6
            e[8] += s2 * s6 * n2;                     // Z2Z6
        }

        for (int off = 16; off > 0; off >>= 1)
            for (int k = 0; k < 9; ++k)
                e[k] += __shfl_xor(e[k], off, 32);

        if (lane == 0) {
            const float inv = 1.0f / ssq[s];
            for (int k = 0; k < 9; ++k)
                out[(b0 + s) * 9 + k] = e[k] * inv;
        }
    }
}

// ---------------------------------------------------------------------------
// Launch
// ---------------------------------------------------------------------------
extern "C" void kernel_launch(void* const* d_in, const int* in_sizes, int n_in,
                              void* d_out, int out_size, void* d_ws, size_t ws_size,
                              hipStream_t stream) {
    const float* x = (const float*)d_in[0];   // (8192, 256) f32
    const float* w = (const float*)d_in[1];   // (75,) f32
    float* out = (float*)d_out;               // (8192, 9) f32

    char* ws = (char*)d_ws;
    cpx*   G   = (cpx*)ws;                             // 640 B
    float* Ure = (float*)(ws + 1024);                  // 256 KB
    float* Uim = (float*)(ws + 1024 + 256 * 256 * 4);  // 256 KB

    build_gates<<<1, 32, 0, stream>>>(w, G);
    build_unitary<<<256, 64, 0, stream>>>(G, Ure, Uim);

    const int batch = in_sizes[0] / 256;               // 8192
    gemm_measure<<<batch / 16, 256, 0, stream>>>(x, Ure, Uim, out);
}